// SVM_71691594105115
// MI455X (gfx1250) — compile-verified
//
#include <hip/hip_runtime.h>

typedef float v2f __attribute__((ext_vector_type(2)));
typedef float v8f __attribute__((ext_vector_type(8)));

#define N_PTS 8192

// workspace layout (float offsets)
#define WS_P      0                    // float4 per point: (x0, x1, sq, 1)  -> 4*N floats
#define WS_BSUM   (4 * N_PTS)          // 32 per-block sums
#define WS_BSQ    (4 * N_PTS + 32)     // 32 per-block sum-of-squares
#define WS_STATS  (4 * N_PTS + 64)     // [0]=mean, [1]=inv_std

// ---------------------------------------------------------------------------
// Stage 1: X = XX @ W1^T + b1 ; per-block partial sums for mean/std
// ---------------------------------------------------------------------------
__global__ void svm_stats_kernel(const float* __restrict__ XX,
                                 const float* __restrict__ W1,
                                 const float* __restrict__ b1,
                                 float* __restrict__ ws) {
    __shared__ float s_sum[256];
    __shared__ float s_sq[256];
    int i = blockIdx.x * 256 + threadIdx.x;
    float xi0 = XX[2 * i], xi1 = XX[2 * i + 1];
    float w00 = W1[0], w01 = W1[1], w10 = W1[2], w11 = W1[3];
    float x0 = fmaf(xi1, w01, fmaf(xi0, w00, b1[0]));
    float x1 = fmaf(xi1, w11, fmaf(xi0, w10, b1[1]));
    s_sum[threadIdx.x] = x0 + x1;
    s_sq[threadIdx.x]  = fmaf(x0, x0, x1 * x1);
    __syncthreads();
    for (int off = 128; off >= 1; off >>= 1) {
        if ((int)threadIdx.x < off) {
            s_sum[threadIdx.x] += s_sum[threadIdx.x + off];
            s_sq[threadIdx.x]  += s_sq[threadIdx.x + off];
        }
        __syncthreads();
    }
    if (threadIdx.x == 0) {
        ws[WS_BSUM + blockIdx.x] = s_sum[0];
        ws[WS_BSQ  + blockIdx.x] = s_sq[0];
    }
}

// ---------------------------------------------------------------------------
// Stage 2: finalize mean / inv_std (std with ddof=1), single wave32
// ---------------------------------------------------------------------------
__global__ void svm_finalize_stats(float* __restrict__ ws) {
    int l = threadIdx.x;  // 0..31
    float s = ws[WS_BSUM + l];
    float q = ws[WS_BSQ + l];
    for (int off = 16; off >= 1; off >>= 1) {
        s += __shfl_xor(s, off, 32);
        q += __shfl_xor(q, off, 32);
    }
    if (l == 0) {
        const float M = 2.0f * (float)N_PTS;   // total element count
        float mean = s / M;
        float var  = (q - s * s / M) / (M - 1.0f);
        ws[WS_STATS + 0] = mean;
        ws[WS_STATS + 1] = rsqrtf(var);        // 1/std
    }
}

// ---------------------------------------------------------------------------
// Stage 3: recompute X, standardize, pack P[i] = (x0, x1, sq, 1) as float4
// ---------------------------------------------------------------------------
__global__ void svm_pack_kernel(const float* __restrict__ XX,
                                const float* __restrict__ W1,
                                const float* __restrict__ b1,
                                float* __restrict__ ws) {
    int i = blockIdx.x * 256 + threadIdx.x;
    float mean = ws[WS_STATS + 0];
    float inv  = ws[WS_STATS + 1];
    float xi0 = XX[2 * i], xi1 = XX[2 * i + 1];
    float w00 = W1[0], w01 = W1[1], w10 = W1[2], w11 = W1[3];
    float x0 = fmaf(xi1, w01, fmaf(xi0, w00, b1[0]));
    float x1 = fmaf(xi1, w11, fmaf(xi0, w10, b1[1]));
    x0 = (x0 - mean) * inv;
    x1 = (x1 - mean) * inv;
    float sq = fmaf(x0, x0, x1 * x1);
    float4* P = (float4*)(ws + WS_P);
    P[i] = make_float4(x0, x1, sq, 1.0f);
}

// ---------------------------------------------------------------------------
// Stage 4: fused pairwise-RBF + weighted row reduction via f32 WMMA.
// One 16-row block per workgroup (8 wave32); each wave handles 64 column
// tiles. K-tile = A(16x4) x B(4x16) with A row i = [x0,x1,sq,1] and
// B col j = [-2x0,-2x1,1,sq]  =>  D = sq_i + sq_j - 2 x_i.x_j.
// ---------------------------------------------------------------------------
__global__ void __launch_bounds__(256, 1)
svm_rbf_wmma_kernel(const float* __restrict__ W2,
                    const float* __restrict__ b2,
                    const float* __restrict__ ws,
                    float* __restrict__ out) {
    const float4* __restrict__ P = (const float4*)(ws + WS_P);

    const int lane = threadIdx.x & 31;
    const int wave = threadIdx.x >> 5;
    const int half = lane >> 4;    // 0: K=0/1 rows, 1: K=2/3 rows of A/B frags
    const int l15  = lane & 15;
    const int ib   = blockIdx.x;   // 16-row block

    // A fragment: 32-bit A 16x4 layout. lanes 0-15: VGPR0=K0, VGPR1=K1;
    // lanes 16-31: VGPR0=K2, VGPR1=K3.  Row = l15.
    float4 pr = P[ib * 16 + l15];
    v2f a;
    a.x = half ? pr.z : pr.x;      // K0 = x0 | K2 = sq
    a.y = half ? pr.w : pr.y;      // K1 = x1 | K3 = 1

    v8f acc = {0.f, 0.f, 0.f, 0.f, 0.f, 0.f, 0.f, 0.f};

    const int nColBlocks = N_PTS / 16;   // 512
    for (int jb = wave; jb < nColBlocks; jb += 8) {
        const int c = jb * 16 + l15;     // column index for this lane (N = l15)
        float4 q = P[c];
        // B fragment: col j = [-2x0, -2x1, 1, sq]; lanes 0-15 hold K0/K1,
        // lanes 16-31 hold K2/K3.
        v2f b;
        b.x = half ? 1.0f : -2.0f * q.x;
        b.y = half ? q.z  : -2.0f * q.y;

        v8f cz = {0.f, 0.f, 0.f, 0.f, 0.f, 0.f, 0.f, 0.f};
        v8f d = __builtin_amdgcn_wmma_f32_16x16x4_f32(
            /*neg_a=*/false, a, /*neg_b=*/false, b,
            /*c_mod=*/(short)0, cz, /*reuse_a=*/false, /*reuse_b=*/false);

        const float wl = -0.5f * W2[c];  // fold gamma=1/2 and column weight
#pragma unroll
        for (int v = 0; v < 8; ++v) {
            float k = fmaxf(d[v], 0.0f);           // relu clamp
            acc[v] = fmaf(wl * k, __expf(k), acc[v]);  // += w*(-0.5k)e^k
        }
    }

    // Reduce across the 16 lanes of each half (C/D layout: N = lane&15,
    // rows v (lanes 0-15) and v+8 (lanes 16-31)). XOR offsets <=8 stay
    // within each 16-lane half.
#pragma unroll
    for (int v = 0; v < 8; ++v) {
        float s = acc[v];
        s += __shfl_xor(s, 1, 32);
        s += __shfl_xor(s, 2, 32);
        s += __shfl_xor(s, 4, 32);
        s += __shfl_xor(s, 8, 32);
        acc[v] = s;
    }

    __shared__ float part[8][16];
    if (l15 == 0) {                      // lane 0 -> rows 0-7, lane 16 -> rows 8-15
#pragma unroll
        for (int v = 0; v < 8; ++v)
            part[wave][half * 8 + v] = acc[v];
    }
    __syncthreads();

    if (threadIdx.x < 16) {
        float s = 0.0f;
#pragma unroll
        for (int w = 0; w < 8; ++w) s += part[w][threadIdx.x];
        out[ib * 16 + threadIdx.x] = s + b2[0];
    }
}

// ---------------------------------------------------------------------------
extern "C" void kernel_launch(void* const* d_in, const int* in_sizes, int n_in,
                              void* d_out, int out_size, void* d_ws, size_t ws_size,
                              hipStream_t stream) {
    const float* XX = (const float*)d_in[0];  // [8192,2]
    const float* W1 = (const float*)d_in[1];  // [2,2]
    const float* b1 = (const float*)d_in[2];  // [2]
    const float* W2 = (const float*)d_in[3];  // [1,8192]
    const float* b2 = (const float*)d_in[4];  // [1]
    float* out = (float*)d_out;               // [8192]
    float* ws  = (float*)d_ws;

    svm_stats_kernel<<<N_PTS / 256, 256, 0, stream>>>(XX, W1, b1, ws);
    svm_finalize_stats<<<1, 32, 0, stream>>>(ws);
    svm_pack_kernel<<<N_PTS / 256, 256, 0, stream>>>(XX, W1, b1, ws);
    svm_rbf_wmma_kernel<<<N_PTS / 16, 256, 0, stream>>>(W2, b2, ws, out);
}